// CCD_CRP_Loss_21440476742411
// MI455X (gfx1250) — compile-verified
//
#include <hip/hip_runtime.h>
#include <hip/hip_bf16.h>
#include <math.h>

typedef __attribute__((ext_vector_type(2))) float v2f;
typedef __attribute__((ext_vector_type(8))) float v8f;

#define B_SZ   64
#define FEAT   128
#define SDIM   2048
#define PK     8196      // num_pos + K = 4 + 8192
#define NCLS   7
#define PER    1000
#define NDATAF 100000.0f
#define TNCE   0.07f
#define EPS_C  1e-07f

// ---- workspace layout (float offsets) ----
static constexpr size_t WS_FS     = 0;                       // f_s normalized (aligned copy)
static constexpr size_t WS_FT     = WS_FS + B_SZ * FEAT;     // f_t normalized
static constexpr size_t WS_UPD_S  = WS_FT + B_SZ * FEAT;     // l2norm momentum-updated rows (s)
static constexpr size_t WS_UPD_T  = WS_UPD_S + B_SZ * FEAT;  // (t)
static constexpr size_t WS_SUM    = WS_UPD_T + B_SZ * FEAT;  // [0]=sum exp s, [1]=sum exp t
static constexpr size_t WS_ANCH_S = WS_SUM + 8;              // 7x128 anchors (s)
static constexpr size_t WS_ANCH_T = WS_ANCH_S + NCLS * FEAT; // 7x128 anchors (t)
static constexpr size_t WS_DOTS_S = WS_ANCH_T + NCLS * FEAT; // 64x8196 raw dots/T (s side)
static constexpr size_t WS_DOTS_T = WS_DOTS_S + (size_t)B_SZ * PK;

// ============================================================================
// Kernel A: fused projection GEMM (64x2048 @ 2048x128) + bias + row L2-norm.
// One block of 1024 threads = 32 waves; wave w owns output tile (mt=w/8, nt=w%8).
// Uses V_WMMA_F32_16X16X4_F32 over K=2048 (512 MACs per wave).
// ============================================================================
__global__ __launch_bounds__(1024)
void proj_l2norm_wmma(const float* __restrict__ X, const float* __restrict__ W,
                      const float* __restrict__ bias, float* __restrict__ out,
                      float* __restrict__ wscopy) {
    __shared__ float tile[B_SZ][FEAT];
    __shared__ float rinv[B_SZ];
    const int tid  = threadIdx.x;
    const int wave = tid >> 5;
    const int lane = tid & 31;
    const int half = lane >> 4;   // which 16-lane half of the wave
    const int l16  = lane & 15;
    const int mt   = wave >> 3;   // 0..3
    const int nt   = wave & 7;    // 0..7
    const int row  = mt * 16 + l16;   // A row (M) held by this lane
    const int col  = nt * 16 + l16;   // B/D column (N) held by this lane

    v8f acc = {0.f, 0.f, 0.f, 0.f, 0.f, 0.f, 0.f, 0.f};
    const float* arow = X + (size_t)row * SDIM;
    for (int k = 0; k < SDIM; k += 4) {
        // A 16x4 layout: VGPR0 = K(2*half), VGPR1 = K(2*half+1), M = lane%16
        // B 4x16 layout: VGPR0 = row K(2*half), VGPR1 = row K(2*half+1), N = lane%16
        const int ka = k + 2 * half;
        v2f a, b;
        a.x = arow[ka];
        a.y = arow[ka + 1];
        b.x = W[(size_t)(ka)     * FEAT + col];
        b.y = W[(size_t)(ka + 1) * FEAT + col];
        acc = __builtin_amdgcn_wmma_f32_16x16x4_f32(
            /*neg_a=*/false, a, /*neg_b=*/false, b,
            /*c_mod=*/(short)0, acc, /*reuse_a=*/false, /*reuse_b=*/false);
    }
    // C/D 16x16 f32 layout: VGPR v -> M = v + 8*half, N = lane%16
#pragma unroll
    for (int v = 0; v < 8; ++v) {
        const int m = mt * 16 + v + 8 * half;
        tile[m][col] = acc[v] + bias[col];
    }
    __syncthreads();
    if (tid < B_SZ) {
        float s = 0.f;
        for (int c = 0; c < FEAT; ++c) { const float x = tile[tid][c]; s += x * x; }
        rinv[tid] = rsqrtf(s);
    }
    __syncthreads();
    for (int e = tid; e < B_SZ * FEAT; e += 1024) {
        const int m = e >> 7, c = e & 127;
        const float val = tile[m][c] * rinv[m];
        out[e]    = val;   // d_out region (possibly unaligned for vectors; scalar store)
        wscopy[e] = val;   // aligned copy for downstream kernels
    }
}

// ============================================================================
// Kernel B: upd = l2norm(0.5*mem[idx[b]] + 0.5*f[b])  (one block per row)
// ============================================================================
__global__ __launch_bounds__(FEAT)
void momentum_update(const float* __restrict__ mem, const float* __restrict__ f,
                     const int* __restrict__ idx, float* __restrict__ upd) {
    __shared__ float red[FEAT];
    const int b = blockIdx.x, d = threadIdx.x;
    const int row = idx[b];
    const float v = 0.5f * mem[(size_t)row * FEAT + d] + 0.5f * f[b * FEAT + d];
    red[d] = v * v;
    __syncthreads();
    for (int s = FEAT / 2; s > 0; s >>= 1) {
        if (d < s) red[d] += red[d + s];
        __syncthreads();
    }
    upd[b * FEAT + d] = v * rsqrtf(red[0]);
}

// ============================================================================
// Kernel C: dominant gather-dot (memory bound, ~268 MB gathered per side).
// One wave per (b,k): dot(mem[contrast_idx[b,k]], f[b]) / T; stores raw dot,
// accumulates sum(exp(dot)) for the NCE normalizer.
// ============================================================================
__global__ __launch_bounds__(256)
void nce_dots(const float* __restrict__ mem, const float* __restrict__ f,
              const int* __restrict__ cidx, float* __restrict__ dots,
              float* __restrict__ sum_accum) {
    __shared__ float fsh[FEAT];
    __shared__ float part[8];
    const int b = blockIdx.y;
    const int tid = threadIdx.x;
    if (tid < FEAT) fsh[tid] = f[b * FEAT + tid];
    __syncthreads();
    const int wave = tid >> 5, lane = tid & 31;
    const int k = blockIdx.x * 8 + wave;
    float e = 0.f;
    if (k < PK) {
        const int row = cidx[(size_t)b * PK + k];
        const float* mrow = mem + (size_t)row * FEAT;
        float d = 0.f;
#pragma unroll
        for (int i = 0; i < FEAT; i += 32) d += mrow[lane + i] * fsh[lane + i];
        for (int off = 16; off > 0; off >>= 1) d += __shfl_down(d, off, 32);
        if (lane == 0) {
            d /= TNCE;
            dots[(size_t)b * PK + k] = d;
            e = expf(d);
        }
    }
    if (lane == 0) part[wave] = e;
    __syncthreads();
    if (tid == 0) {
        const float s = part[0] + part[1] + part[2] + part[3] +
                        part[4] + part[5] + part[6] + part[7];
        atomicAdd(sum_accum, s);
    }
}

// ============================================================================
// Kernel D: CCD contrastive loss for both sides.
//   x = exp(dot)/Z, Z = mean(exp)*N_DATA
//   loss = -[sum_{k<P} log(x/(x+m+eps))]/(B*P) - [sum_{k>=P} log(m/(x+m+eps))]/B
// ============================================================================
__global__ __launch_bounds__(256)
void ccd_loss(const float* __restrict__ dots_s, const float* __restrict__ dots_t,
              const float* __restrict__ sums, const int* __restrict__ num_pos,
              float* __restrict__ out_loss) {
    const int P = num_pos[0];
    const float m_pn = (float)(PK - P) / NDATAF;
    const float Zs = (sums[0] / (float)(B_SZ * PK)) * NDATAF;
    const float Zt = (sums[1] / (float)(B_SZ * PK)) * NDATAF;
    const size_t per_side = (size_t)B_SZ * PK;
    const size_t total = 2 * per_side;
    float local = 0.f;
    for (size_t i = (size_t)blockIdx.x * blockDim.x + threadIdx.x; i < total;
         i += (size_t)gridDim.x * blockDim.x) {
        const int side = (int)(i / per_side);
        const size_t j = i % per_side;
        const int k = (int)(j % PK);
        const float dot = side ? dots_t[j] : dots_s[j];
        const float Z = side ? Zt : Zs;
        const float x = expf(dot) / Z;
        if (k < P) local += -logf(x / (x + m_pn + EPS_C)) / (float)(B_SZ * P);
        else       local += -logf(m_pn / (x + m_pn + EPS_C)) / (float)B_SZ;
    }
    __shared__ float red[256];
    red[threadIdx.x] = local;
    __syncthreads();
    for (int s = 128; s > 0; s >>= 1) {
        if (threadIdx.x < s) red[threadIdx.x] += red[threadIdx.x + s];
        __syncthreads();
    }
    if (threadIdx.x == 0) atomicAdd(out_loss, red[0]);
}

// ============================================================================
// Kernel E: class anchors from *updated* memory, without materializing it.
// map[j] = last b with idx[b]==class_index[c][j] (JAX scatter: last write wins),
// anchors[c] = l2norm(mean_j relu(row_j)).
// ============================================================================
__global__ __launch_bounds__(FEAT)
void anchors_kernel(const float* __restrict__ mem, const float* __restrict__ upd,
                    const int* __restrict__ class_index, const int* __restrict__ idx,
                    float* __restrict__ anchors) {
    __shared__ int map[PER];
    __shared__ float red[FEAT];
    const int c = blockIdx.x, d = threadIdx.x;
    for (int j = d; j < PER; j += FEAT) {
        const int row = class_index[c * PER + j];
        int m = -1;
        for (int b = 0; b < B_SZ; ++b)
            if (idx[b] == row) m = b;
        map[j] = m;
    }
    __syncthreads();
    float s = 0.f;
    for (int j = 0; j < PER; ++j) {
        const int row = class_index[c * PER + j];
        const float v = (map[j] >= 0) ? upd[map[j] * FEAT + d]
                                      : mem[(size_t)row * FEAT + d];
        s += fmaxf(v, 0.f);
    }
    const float mean = s / (float)PER;
    red[d] = mean * mean;
    __syncthreads();
    for (int st = FEAT / 2; st > 0; st >>= 1) {
        if (d < st) red[d] += red[d + st];
        __syncthreads();
    }
    anchors[c * FEAT + d] = mean * rsqrtf(red[0]);
}

// ============================================================================
// Kernel F: relation (KL) loss. 64x7 logits per side; softmax targets from t,
// log-softmax inputs from s. One thread per batch row.
// ============================================================================
__global__ __launch_bounds__(B_SZ)
void relation_kernel(const float* __restrict__ fs, const float* __restrict__ ft,
                     const float* __restrict__ as, const float* __restrict__ at,
                     float* __restrict__ out_rel) {
    __shared__ float red[B_SZ];
    const int b = threadIdx.x;
    float srel[NCLS], trel[NCLS];
    for (int c = 0; c < NCLS; ++c) {
        float ds = 0.f, dt = 0.f;
        for (int d = 0; d < FEAT; ++d) {
            ds += fs[b * FEAT + d] * as[c * FEAT + d];
            dt += ft[b * FEAT + d] * at[c * FEAT + d];
        }
        srel[c] = ds / TNCE;
        trel[c] = dt / TNCE;
    }
    float mt = trel[0];
    for (int c = 1; c < NCLS; ++c) mt = fmaxf(mt, trel[c]);
    float tg[NCLS]; float st = 0.f;
    for (int c = 0; c < NCLS; ++c) { tg[c] = expf(trel[c] - mt); st += tg[c]; }
    for (int c = 0; c < NCLS; ++c) tg[c] /= st;
    float ms = srel[0];
    for (int c = 1; c < NCLS; ++c) ms = fmaxf(ms, srel[c]);
    float ss = 0.f;
    for (int c = 0; c < NCLS; ++c) ss += expf(srel[c] - ms);
    const float lse = ms + logf(ss);
    float l = 0.f;
    for (int c = 0; c < NCLS; ++c)
        l += tg[c] * (logf(tg[c] + 1e-30f) - (srel[c] - lse));
    red[b] = l;
    __syncthreads();
    for (int s = 32; s > 0; s >>= 1) {
        if (b < s) red[b] += red[b + s];
        __syncthreads();
    }
    if (b == 0) out_rel[0] = red[0] / (float)B_SZ;
}

// ============================================================================
extern "C" void kernel_launch(void* const* d_in, const int* in_sizes, int n_in,
                              void* d_out, int out_size, void* d_ws, size_t ws_size,
                              hipStream_t stream) {
    const float* f_s_in      = (const float*)d_in[0];
    const float* f_t_in      = (const float*)d_in[1];
    const int*   idx         = (const int*)d_in[2];
    /* d_in[3] batch_label: unused by the reference */
    const int*   class_index = (const int*)d_in[4];
    const int*   num_pos     = (const int*)d_in[5];
    const int*   contrast_idx= (const int*)d_in[6];
    const float* W_s         = (const float*)d_in[7];
    const float* b_s         = (const float*)d_in[8];
    const float* W_t         = (const float*)d_in[9];
    const float* b_t         = (const float*)d_in[10];
    const float* memory_s    = (const float*)d_in[11];
    const float* memory_t    = (const float*)d_in[12];

    float* out = (float*)d_out;                 // [ccd, rel, f_s(8192), f_t(8192)]
    float* ws  = (float*)d_ws;
    float* fs_out = out + 2;
    float* ft_out = out + 2 + B_SZ * FEAT;

    // zero accumulators (memset nodes are graph-capture safe)
    hipMemsetAsync(out, 0, 2 * sizeof(float), stream);
    hipMemsetAsync(ws + WS_SUM, 0, 2 * sizeof(float), stream);

    // 1) projections + l2norm (WMMA f32 16x16x4)
    proj_l2norm_wmma<<<1, 1024, 0, stream>>>(f_s_in, W_s, b_s, fs_out, ws + WS_FS);
    proj_l2norm_wmma<<<1, 1024, 0, stream>>>(f_t_in, W_t, b_t, ft_out, ws + WS_FT);

    // 2) momentum memory updates (virtual scatter rows)
    momentum_update<<<B_SZ, FEAT, 0, stream>>>(memory_s, ws + WS_FS, idx, ws + WS_UPD_S);
    momentum_update<<<B_SZ, FEAT, 0, stream>>>(memory_t, ws + WS_FT, idx, ws + WS_UPD_T);

    // 3) NCE gather-dots (dominant, memory-bound) + exp-sum for normalizer
    dim3 gdots((PK + 7) / 8, B_SZ, 1);
    nce_dots<<<gdots, 256, 0, stream>>>(memory_t, ws + WS_FS, contrast_idx,
                                        ws + WS_DOTS_S, ws + WS_SUM + 0);
    nce_dots<<<gdots, 256, 0, stream>>>(memory_s, ws + WS_FT, contrast_idx,
                                        ws + WS_DOTS_T, ws + WS_SUM + 1);

    // 4) CCD contrastive loss (both sides) -> out[0]
    ccd_loss<<<256, 256, 0, stream>>>(ws + WS_DOTS_S, ws + WS_DOTS_T,
                                      ws + WS_SUM, num_pos, out + 0);

    // 5) class anchors from updated memory
    anchors_kernel<<<NCLS, FEAT, 0, stream>>>(memory_s, ws + WS_UPD_S, class_index, idx, ws + WS_ANCH_S);
    anchors_kernel<<<NCLS, FEAT, 0, stream>>>(memory_t, ws + WS_UPD_T, class_index, idx, ws + WS_ANCH_T);

    // 6) relation (KL) loss -> out[1]
    relation_kernel<<<1, B_SZ, 0, stream>>>(ws + WS_FS, ws + WS_FT,
                                            ws + WS_ANCH_S, ws + WS_ANCH_T, out + 1);
}